// ContrastiveLoss_55774445306458
// MI455X (gfx1250) — compile-verified
//
#include <hip/hip_runtime.h>
#include <hip/hip_bf16.h>
#include <math.h>

typedef __attribute__((ext_vector_type(2))) float v2f;
typedef __attribute__((ext_vector_type(8))) float v8f;

#define B_SZ 192
#define N_SZ 384
#define D_SZ 128
#define TEMP_INV 0.5f   // 1/TEMP, TEMP=2.0
// TAU = 1.0, POS_W = 0.1, NEG_W = 1.0, SOFT = true

// ---------------------------------------------------------------------------
// Phase 0: pack z = transpose(emb,(1,0,2)).reshape(N,D) into contiguous [N,D]
// emb is [192,2,128] row major; z[r][d] = emb[r%192][r/192][d]
// ---------------------------------------------------------------------------
__global__ void pack_z_kernel(const float* __restrict__ emb, float* __restrict__ Zc) {
    int r = blockIdx.x;
    int d = threadIdx.x;
    int b = r % B_SZ;
    int c = r / B_SZ;
    Zc[r * D_SZ + d] = emb[b * (2 * D_SZ) + c * D_SZ + d];
}

// ---------------------------------------------------------------------------
// Phase 1: Gram matrix G = Z * Z^T via V_WMMA_F32_16X16X4_F32 (full fp32).
// One 16x16 output tile per wave; 24x24 tiles; K-loop 128/4 = 32 WMMAs.
// A 16x4 f32 layout: lane<16 -> row=lane, {K=k,k+1}; lane>=16 -> row=lane-16,
// {K=k+2,k+3}. B 4x16 mirrors it with lane = column. C/D: VGPR v holds
// (M=v, N=lane) for lanes 0-15 and (M=v+8, N=lane-16) for lanes 16-31.
// ---------------------------------------------------------------------------
__global__ void gram_wmma_kernel(const float* __restrict__ Zc, float* __restrict__ G) {
    const int wid  = blockIdx.x * (blockDim.x >> 5) + (threadIdx.x >> 5);
    const int lane = threadIdx.x & 31;
    const int ti = wid / (N_SZ / 16);
    const int tj = wid % (N_SZ / 16);
    const int ibase = ti * 16;
    const int jbase = tj * 16;
    const int hl = lane >> 4;     // 0 or 1 (lane half)
    const int lr = lane & 15;

    const float* __restrict__ Ar = Zc + (ibase + lr) * D_SZ + 2 * hl;
    const float* __restrict__ Br = Zc + (jbase + lr) * D_SZ + 2 * hl;

    v8f acc = {};
    for (int k = 0; k < D_SZ; k += 4) {
        v2f a, b;
        a.x = Ar[k];     a.y = Ar[k + 1];
        b.x = Br[k];     b.y = Br[k + 1];
        // 8 args: (neg_a, A, neg_b, B, c_mod, C, reuse_a, reuse_b)
        acc = __builtin_amdgcn_wmma_f32_16x16x4_f32(
            false, a, false, b, (short)0, acc, false, false);
    }
#pragma unroll
    for (int v = 0; v < 8; ++v) {
        int m = v + 8 * hl;
        G[(ibase + m) * N_SZ + (jbase + lr)] = acc[v];
    }
}

// ---------------------------------------------------------------------------
// Phase 2: per-pair terms.
//   sq   = G_ii + G_jj - 2 G_ij (clamped >= 0), dist = sqrt(sq)
//   s    = -dist/TEMP
//   sg   = y_i - y_j, ab = |sg|, dw = sigmoid(ab)
//   w    = (i==j) ? 0 : exp(s)*dw
//   cw   = w * (sg<0 ? POS_W : 0)   [mask value when j is strictly closer]
// ---------------------------------------------------------------------------
__global__ void pairwise_kernel(const float* __restrict__ G,
                                const float* __restrict__ targets,
                                float* __restrict__ S, float* __restrict__ W,
                                float* __restrict__ CW, float* __restrict__ AB) {
    int i = blockIdx.x;
    int j = blockIdx.y * blockDim.x + threadIdx.x;

    float g  = G[i * N_SZ + j];
    float sq = G[i * N_SZ + i] + G[j * N_SZ + j] - 2.0f * g;
    sq = fmaxf(sq, 0.0f);
    float dist = sqrtf(sq);
    float s = -dist * TEMP_INV;

    float yi = targets[i % B_SZ];
    float yj = targets[j % B_SZ];
    float sg = yi - yj;
    float ab = fabsf(sg);
    float dw = 1.0f / (1.0f + expf(-ab));      // sigmoid(TAU*ab), TAU=1

    float w  = (i == j) ? 0.0f : expf(s) * dw;
    float cw = w * ((sg < 0.0f) ? 0.1f : 0.0f);

    int idx = i * N_SZ + j;
    S[idx]  = s;
    W[idx]  = w;
    CW[idx] = cw;
    AB[idx] = ab;
}

// ---------------------------------------------------------------------------
// Phase 3: one block per row i. Cache row (w, cw, ab) in LDS; thread p
// computes denom[i,p] = sum_j (ab_j < ab_p ? cw_j : w_j), then
// contrib = log(denom) - s_ip. Fixed-order LDS tree reduce -> partial[i].
// ---------------------------------------------------------------------------
__global__ void denom_loss_kernel(const float* __restrict__ S,
                                  const float* __restrict__ W,
                                  const float* __restrict__ CW,
                                  const float* __restrict__ AB,
                                  float* __restrict__ partial) {
    __shared__ float wr[N_SZ], cwr[N_SZ], abr[N_SZ], red[512];
    const int i = blockIdx.x;
    const int t = threadIdx.x;

    for (int j = t; j < N_SZ; j += 512) {
        wr[j]  = W[i * N_SZ + j];
        cwr[j] = CW[i * N_SZ + j];
        abr[j] = AB[i * N_SZ + j];
    }
    __syncthreads();

    float contrib = 0.0f;
    if (t < N_SZ && t != i) {
        float abp = abr[t];
        float acc = 0.0f;
        for (int j = 0; j < N_SZ; ++j)
            acc += (abr[j] < abp) ? cwr[j] : wr[j];
        contrib = logf(acc) - S[i * N_SZ + t];
    }
    red[t] = contrib;
    __syncthreads();
    for (int step = 256; step > 0; step >>= 1) {
        if (t < step) red[t] += red[t + step];
        __syncthreads();
    }
    if (t == 0) partial[i] = red[0];
}

// ---------------------------------------------------------------------------
// Phase 4: final deterministic reduce of 384 partials, apply 1/(N*(N-1)).
// ---------------------------------------------------------------------------
__global__ void final_kernel(const float* __restrict__ partial,
                             float* __restrict__ out) {
    __shared__ float red[512];
    const int t = threadIdx.x;
    red[t] = (t < N_SZ) ? partial[t] : 0.0f;
    __syncthreads();
    for (int step = 256; step > 0; step >>= 1) {
        if (t < step) red[t] += red[t + step];
        __syncthreads();
    }
    if (t == 0) out[0] = red[0] / (float)(N_SZ * (N_SZ - 1));
}

// ---------------------------------------------------------------------------
extern "C" void kernel_launch(void* const* d_in, const int* in_sizes, int n_in,
                              void* d_out, int out_size, void* d_ws, size_t ws_size,
                              hipStream_t stream) {
    const float* emb = (const float*)d_in[0];   // [192, 2, 128] fp32
    const float* tgt = (const float*)d_in[1];   // [192, 1] fp32

    float* ws = (float*)d_ws;
    float* Zc      = ws;                         // 384*128  = 49152
    float* G       = Zc + N_SZ * D_SZ;           // 384*384  = 147456
    float* S       = G  + N_SZ * N_SZ;           // 147456
    float* W       = S  + N_SZ * N_SZ;           // 147456
    float* CW      = W  + N_SZ * N_SZ;           // 147456
    float* AB      = CW + N_SZ * N_SZ;           // 147456
    float* partial = AB + N_SZ * N_SZ;           // 384
    // total ~3.15 MB of workspace

    pack_z_kernel<<<N_SZ, D_SZ, 0, stream>>>(emb, Zc);

    // 24*24 = 576 tiles, 8 waves (256 threads) per block -> 72 blocks
    gram_wmma_kernel<<<(24 * 24) / 8, 256, 0, stream>>>(Zc, G);

    dim3 g2(N_SZ, N_SZ / 128);
    pairwise_kernel<<<g2, 128, 0, stream>>>(G, tgt, S, W, CW, AB);

    denom_loss_kernel<<<N_SZ, 512, 0, stream>>>(S, W, CW, AB, partial);

    final_kernel<<<1, 512, 0, stream>>>(partial, (float*)d_out);
}